// StandardGATModel_4380866642061
// MI455X (gfx1250) — compile-verified
//
#include <hip/hip_runtime.h>
#include <hip/hip_bf16.h>
#include <math.h>

typedef __attribute__((ext_vector_type(2))) float v2f;
typedef __attribute__((ext_vector_type(8))) float v8f;

#define N_NODES 50000
#define N_EDGES 800000
#define E_TOT   (N_EDGES + N_NODES)

// ---------------------------------------------------------------- utilities
__global__ void fill_kernel(float* __restrict__ p, float v, int n) {
    int i = blockIdx.x * blockDim.x + threadIdx.x;
    if (i < n) p[i] = v;
}

__device__ __forceinline__ void edge_nodes(const int* __restrict__ ei, int e, int& s, int& d) {
    if (e < N_EDGES) { s = ei[e]; d = ei[N_EDGES + e]; }
    else             { s = d = e - N_EDGES; }   // appended self-loops
}

__device__ __forceinline__ float lrelu(float x) { return x >= 0.f ? x : 0.2f * x; }

// float atomic max via monotone int reinterpretation (lowers to native
// GLOBAL_ATOMIC_MAX_I32 / GLOBAL_ATOMIC_MIN_U32)
__device__ __forceinline__ void atomicMaxF32(float* addr, float val) {
    if (val >= 0.f) atomicMax((int*)addr, __float_as_int(val));
    else            atomicMin((unsigned int*)addr, __float_as_uint(val));
}

// ---------------------------------------------------------------- GEMM (WMMA fp32)
// C[M,Nc] = A[M,K] @ B[K,Nc], row-major fp32 via V_WMMA_F32_16X16X4_F32.
// Requires M%16==0, K%4==0, Nc%64==0. One wave computes a 16x64 tile.
__global__ void __launch_bounds__(256)
gemm_wmma_f32(const float* __restrict__ A, const float* __restrict__ B,
              float* __restrict__ C, int M, int K, int Nc) {
    const int lane = threadIdx.x & 31;
    const int wave = threadIdx.x >> 5;
    const int mt   = blockIdx.x * 8 + wave;        // 16-row tile index (wave-uniform)
    if (mt * 16 >= M) return;                      // whole-wave exit: EXEC stays all-1s
    const int nt   = blockIdx.y;                   // 64-col tile index

    const int row  = mt * 16 + (lane & 15);        // A row for this lane
    const int koff = (lane >> 4) * 2;              // lanes 0-15: K=0,1  lanes 16-31: K=2,3
    const int col0 = nt * 64 + (lane & 15);        // B/C column for this lane

    v8f c0 = {}, c1 = {}, c2 = {}, c3 = {};
    for (int k = 0; k < K; k += 4) {
        v2f a;
        a.x = A[(size_t)row * K + k + koff];
        a.y = A[(size_t)row * K + k + koff + 1];
        const float* Bk = B + (size_t)(k + koff) * Nc + col0;
        v2f b0v, b1v, b2v, b3v;
        b0v.x = Bk[0];   b0v.y = Bk[Nc];
        b1v.x = Bk[16];  b1v.y = Bk[Nc + 16];
        b2v.x = Bk[32];  b2v.y = Bk[Nc + 32];
        b3v.x = Bk[48];  b3v.y = Bk[Nc + 48];
        c0 = __builtin_amdgcn_wmma_f32_16x16x4_f32(false, a, false, b0v, (short)0, c0, false, false);
        c1 = __builtin_amdgcn_wmma_f32_16x16x4_f32(false, a, false, b1v, (short)0, c1, false, false);
        c2 = __builtin_amdgcn_wmma_f32_16x16x4_f32(false, a, false, b2v, (short)0, c2, false, false);
        c3 = __builtin_amdgcn_wmma_f32_16x16x4_f32(false, a, false, b3v, (short)0, c3, false, false);
    }
    // C/D layout: VGPR i -> row = tile*16 + i + 8*(lane>=16), col = col0
    const int rbase = mt * 16 + (lane >> 4) * 8;
#pragma unroll
    for (int i = 0; i < 8; ++i) {
        float* Cr = C + (size_t)(rbase + i) * Nc + col0;
        Cr[0]  = c0[i];
        Cr[16] = c1[i];
        Cr[32] = c2[i];
        Cr[48] = c3[i];
    }
}

// ---------------------------------------------------------------- attention
// asrc[n,h] = sum_c h[n,h,c]*a_src[h,c]; adst analogous
__global__ void scores_kernel(const float* __restrict__ hW,
                              const float* __restrict__ a_src, const float* __restrict__ a_dst,
                              float* __restrict__ asrc, float* __restrict__ adst,
                              int H, int C) {
    int i = blockIdx.x * blockDim.x + threadIdx.x;   // n*H + h
    if (i >= N_NODES * H) return;
    int h = i % H, n = i / H;
    const float* hp = hW + (size_t)n * H * C + (size_t)h * C;
    const float* as = a_src + (size_t)h * C;
    const float* ad = a_dst + (size_t)h * C;
    float ss = 0.f, sd = 0.f;
    for (int c = 0; c < C; ++c) { float v = hp[c]; ss += v * as[c]; sd += v * ad[c]; }
    asrc[i] = ss; adst[i] = sd;
}

__global__ void edge_max_kernel(const int* __restrict__ ei,
                                const float* __restrict__ asrc, const float* __restrict__ adst,
                                float* __restrict__ m, int H) {
    int e = blockIdx.x * blockDim.x + threadIdx.x;
    if (e >= E_TOT) return;
    int s, d; edge_nodes(ei, e, s, d);
    for (int h = 0; h < H; ++h) {
        float l = lrelu(asrc[s * H + h] + adst[d * H + h]);
        atomicMaxF32(&m[d * H + h], l);
    }
}

__global__ void edge_sum_kernel(const int* __restrict__ ei,
                                const float* __restrict__ asrc, const float* __restrict__ adst,
                                const float* __restrict__ m, float* __restrict__ den, int H) {
    int e = blockIdx.x * blockDim.x + threadIdx.x;
    if (e >= E_TOT) return;
    int s, d; edge_nodes(ei, e, s, d);
    for (int h = 0; h < H; ++h) {
        float l = lrelu(asrc[s * H + h] + adst[d * H + h]);
        atomicAdd(&den[d * H + h], expf(l - m[d * H + h]));
    }
}

__global__ void edge_scatter_kernel(const int* __restrict__ ei,
                                    const float* __restrict__ asrc, const float* __restrict__ adst,
                                    const float* __restrict__ m, const float* __restrict__ den,
                                    const float* __restrict__ hW, float* __restrict__ accum,
                                    int H, int C) {
    int i = blockIdx.x * blockDim.x + threadIdx.x;   // e*H + h  (E_TOT*H < 2^31)
    if (i >= E_TOT * H) return;
    int h = i % H, e = i / H;
    int s, d; edge_nodes(ei, e, s, d);
    float l = lrelu(asrc[s * H + h] + adst[d * H + h]);
    float alpha = expf(l - m[d * H + h]) / (den[d * H + h] + 1e-16f);
    const float* hp = hW + (size_t)s * H * C + (size_t)h * C;
    float* op = accum + (size_t)d * H * C + (size_t)h * C;
#pragma unroll 4
    for (int c = 0; c < C; ++c) atomicAdd(&op[c], alpha * hp[c]);
}

__global__ void finalize_kernel(const float* __restrict__ acc, const float* __restrict__ bias,
                                float* __restrict__ out, int HC, int do_elu) {
    int i = blockIdx.x * blockDim.x + threadIdx.x;
    if (i >= N_NODES * HC) return;
    int j = i % HC;
    float v = acc[i] + bias[j];
    if (do_elu) v = v > 0.f ? v : (expf(v) - 1.f);
    out[i] = v;
}

// ---------------------------------------------------------------- host side
static void run_attention(const float* hW, const float* a_s, const float* a_d,
                          int H, int C, float* accum, const int* ei,
                          float* asrc, float* adst, float* m, float* den,
                          hipStream_t stream) {
    const int NH  = N_NODES * H;
    const int NHC = N_NODES * H * C;
    scores_kernel<<<(NH + 255) / 256, 256, 0, stream>>>(hW, a_s, a_d, asrc, adst, H, C);
    fill_kernel<<<(NH + 255) / 256, 256, 0, stream>>>(m, -INFINITY, NH);
    fill_kernel<<<(NH + 255) / 256, 256, 0, stream>>>(den, 0.f, NH);
    fill_kernel<<<(NHC + 255) / 256, 256, 0, stream>>>(accum, 0.f, NHC);
    edge_max_kernel<<<(E_TOT + 255) / 256, 256, 0, stream>>>(ei, asrc, adst, m, H);
    edge_sum_kernel<<<(E_TOT + 255) / 256, 256, 0, stream>>>(ei, asrc, adst, m, den, H);
    edge_scatter_kernel<<<(E_TOT * H + 255) / 256, 256, 0, stream>>>(ei, asrc, adst, m, den, hW, accum, H, C);
}

extern "C" void kernel_launch(void* const* d_in, const int* in_sizes, int n_in,
                              void* d_out, int out_size, void* d_ws, size_t ws_size,
                              hipStream_t stream) {
    const float* x   = (const float*)d_in[0];
    const int*   ei  = (const int*)d_in[1];
    const float* W0  = (const float*)d_in[2];
    const float* as0 = (const float*)d_in[3];
    const float* ad0 = (const float*)d_in[4];
    const float* b0  = (const float*)d_in[5];
    const float* W1  = (const float*)d_in[6];
    const float* as1 = (const float*)d_in[7];
    const float* ad1 = (const float*)d_in[8];
    const float* b1  = (const float*)d_in[9];
    const float* W2  = (const float*)d_in[10];
    const float* as2 = (const float*)d_in[11];
    const float* ad2 = (const float*)d_in[12];
    const float* b2  = (const float*)d_in[13];
    float* out = (float*)d_out;

    // workspace layout (floats)
    float* P    = (float*)d_ws;                      // N*256
    float* Q    = P + (size_t)N_NODES * 256;         // N*256
    float* asrc = Q + (size_t)N_NODES * 256;         // N*4
    float* adst = asrc + (size_t)N_NODES * 4;        // N*4
    float* m    = adst + (size_t)N_NODES * 4;        // N*4
    float* den  = m    + (size_t)N_NODES * 4;        // N*4

    const int MT_BLOCKS = (3125 + 7) / 8;            // 50000/16 row tiles, 8 waves/block

    // ---- layer 0: x[N,64] @ W0[64,256] -> P; attention -> Q; bias+ELU -> P
    gemm_wmma_f32<<<dim3(MT_BLOCKS, 4), 256, 0, stream>>>(x, W0, P, N_NODES, 64, 256);
    run_attention(P, as0, ad0, 4, 64, Q, ei, asrc, adst, m, den, stream);
    finalize_kernel<<<(N_NODES * 256 + 255) / 256, 256, 0, stream>>>(Q, b0, P, 256, 1);

    // ---- layer 1: P[N,256] @ W1[256,256] -> Q; attention -> P; bias+ELU -> Q
    gemm_wmma_f32<<<dim3(MT_BLOCKS, 4), 256, 0, stream>>>(P, W1, Q, N_NODES, 256, 256);
    run_attention(Q, as1, ad1, 4, 64, P, ei, asrc, adst, m, den, stream);
    finalize_kernel<<<(N_NODES * 256 + 255) / 256, 256, 0, stream>>>(P, b1, Q, 256, 1);

    // ---- layer 2: Q[N,256] @ W2[256,64] -> P; attention -> d_out; +bias (no ELU)
    gemm_wmma_f32<<<dim3(MT_BLOCKS, 1), 256, 0, stream>>>(Q, W2, P, N_NODES, 256, 64);
    run_attention(P, as2, ad2, 1, 64, out, ei, asrc, adst, m, den, stream);
    finalize_kernel<<<(N_NODES * 64 + 255) / 256, 256, 0, stream>>>(out, b2, out, 64, 0);
}